// GAT_Custom_26001732010347
// MI455X (gfx1250) — compile-verified
//
#include <hip/hip_runtime.h>
#include <math.h>

// ---------------------------------------------------------------------------
// 2-layer GAT (PyG GATConv semantics) for gfx1250.
//   GEMMs:  V_WMMA_F32_16X16X4_F32, one wave per 16x64 output strip
//           (4 accumulators, A fragment loaded once per k-step as float2).
//   Edge phase: segment softmax via float atomic max/add, scatter via
//           global_atomic_add_f32 (L2-resident on 192MB L2).
// ---------------------------------------------------------------------------

typedef __attribute__((ext_vector_type(2))) float v2f;
typedef __attribute__((ext_vector_type(8))) float v8f;

#define GAT_NEG_SLOPE 0.2f

#if defined(__AMDGCN__) && __has_builtin(__builtin_amdgcn_wmma_f32_16x16x4_f32)
#define HAVE_WMMA_F32 1
#else
#define HAVE_WMMA_F32 0
#endif

// ----------------------------- GEMM (WMMA) ---------------------------------
// D[M x 64] = A[M x K] @ B[K x 64], row-major, M % 16 == 0, F fixed at 64.
// One wave per 16-row strip: 4 x (16x16) accumulators share one A fragment.
// EXEC is all-ones inside the wave (early-out is wave-uniform).
__global__ __launch_bounds__(256)
void gemm_wmma_f32_strip(const float* __restrict__ A, const float* __restrict__ B,
                         float* __restrict__ D, int M, int K)
{
    constexpr int F = 64;
    const int lane = threadIdx.x & 31;
    const int wave = blockIdx.x * (blockDim.x >> 5) + (threadIdx.x >> 5);
    if (wave * 16 >= M) return;                  // wave-uniform

    const int half = lane >> 4;                  // 0 or 1
    const int idx  = lane & 15;

    v8f acc0 = {}, acc1 = {}, acc2 = {}, acc3 = {};

#if HAVE_WMMA_F32
    // A fragment (ISA 16x4 f32 layout): lane holds A[row][k+kb .. k+kb+1],
    // row = lane&15, kb = (lane>>4)*2.  Contiguous pair -> global_load_b64.
    // B fragment (mirror): lane holds B[k+kb][col], B[k+kb+1][col].
    const int kb = half * 2;
    const float* Arow = A + (size_t)(wave * 16 + idx) * K;
    for (int k = 0; k < K; k += 4) {
        const v2f av = *(const v2f*)(Arow + k + kb);
        const float* B0 = B + (size_t)(k + kb) * F + idx;   // row k+kb
        const float* B1 = B0 + F;                            // row k+kb+1
        v2f b0, b1, b2, b3;
        b0.x = B0[0];  b0.y = B1[0];
        b1.x = B0[16]; b1.y = B1[16];
        b2.x = B0[32]; b2.y = B1[32];
        b3.x = B0[48]; b3.y = B1[48];
        acc0 = __builtin_amdgcn_wmma_f32_16x16x4_f32(false, av, false, b0,
                                                     (short)0, acc0, false, false);
        acc1 = __builtin_amdgcn_wmma_f32_16x16x4_f32(false, av, false, b1,
                                                     (short)0, acc1, false, false);
        acc2 = __builtin_amdgcn_wmma_f32_16x16x4_f32(false, av, false, b2,
                                                     (short)0, acc2, false, false);
        acc3 = __builtin_amdgcn_wmma_f32_16x16x4_f32(false, av, false, b3,
                                                     (short)0, acc3, false, false);
    }
#else
    for (int i = 0; i < 8; ++i) {
        const int row = wave * 16 + half * 8 + i;
        float s0 = 0.f, s1 = 0.f, s2 = 0.f, s3 = 0.f;
        for (int k = 0; k < K; ++k) {
            const float a = A[(size_t)row * K + k];
            const float* Bk = B + (size_t)k * F + idx;
            s0 += a * Bk[0];  s1 += a * Bk[16];
            s2 += a * Bk[32]; s3 += a * Bk[48];
        }
        acc0[i] = s0; acc1[i] = s1; acc2[i] = s2; acc3[i] = s3;
    }
#endif
    // C/D layout: lane = col (lane&15), VGPR i -> row i + (lane>>4)*8
#pragma unroll
    for (int i = 0; i < 8; ++i) {
        const int row = wave * 16 + half * 8 + i;
        float* p = D + (size_t)row * F + idx;
        p[0]  = acc0[i];
        p[16] = acc1[i];
        p[32] = acc2[i];
        p[48] = acc3[i];
    }
}

// ------------------------- per-node attention coeffs -----------------------
// a_s[n,h] = dot(xW[n,h,:], att_src[h,:]);  a_d analog.  One thread / (n,h).
__global__ void node_attn(const float* __restrict__ xW,
                          const float* __restrict__ att_src,
                          const float* __restrict__ att_dst,
                          int N, int H, int C,
                          float* __restrict__ a_s, float* __restrict__ a_d)
{
    const int t = blockIdx.x * blockDim.x + threadIdx.x;
    if (t >= N * H) return;
    const int n = t / H;
    const int h = t - n * H;
    const float* xp = xW + (size_t)n * H * C + (size_t)h * C;
    const float* as = att_src + (size_t)h * C;
    const float* ad = att_dst + (size_t)h * C;
    float ss = 0.f, sd = 0.f;
    for (int c = 0; c < C; ++c) {
        const float v = xp[c];
        ss += v * as[c];
        sd += v * ad[c];
    }
    a_s[t] = ss;
    a_d[t] = sd;
}

// --------------------------------- fill ------------------------------------
__global__ void fill_f32(float* __restrict__ p, float v, int n)
{
    const int t = blockIdx.x * blockDim.x + threadIdx.x;
    if (t < n) p[t] = v;
}

// ------------------------- float atomic max (sign split) -------------------
__device__ __forceinline__ void atomicMaxFloat(float* addr, float val)
{
    if (!(val < 0.0f)) {
        atomicMax((int*)addr, __float_as_int(val));
    } else {
        atomicMin((unsigned int*)addr, __float_as_uint(val));
    }
}

__device__ __forceinline__ float leaky(float v)
{
    return v > 0.f ? v : GAT_NEG_SLOPE * v;
}

// edges e < E come from edge_index; e >= E are the implicit self-loops.
__device__ __forceinline__ void edge_nodes(const int* __restrict__ src,
                                           const int* __restrict__ dst,
                                           int E, int e, int& s, int& d)
{
    if (e < E) { s = src[e]; d = dst[e]; }
    else       { s = e - E;  d = e - E; }
}

// ----------------------------- segment max ---------------------------------
__global__ void edge_max(const int* __restrict__ src, const int* __restrict__ dst,
                         int E, int N, int H,
                         const float* __restrict__ a_s, const float* __restrict__ a_d,
                         float* __restrict__ m)
{
    const int e = blockIdx.x * blockDim.x + threadIdx.x;
    if (e >= E + N) return;
    int s, d;
    edge_nodes(src, dst, E, e, s, d);
    for (int h = 0; h < H; ++h) {
        const float v = leaky(a_s[(size_t)s * H + h] + a_d[(size_t)d * H + h]);
        atomicMaxFloat(&m[(size_t)d * H + h], v);
    }
}

// ----------------------------- segment expsum ------------------------------
__global__ void edge_sum(const int* __restrict__ src, const int* __restrict__ dst,
                         int E, int N, int H,
                         const float* __restrict__ a_s, const float* __restrict__ a_d,
                         const float* __restrict__ m, float* __restrict__ denom)
{
    const int e = blockIdx.x * blockDim.x + threadIdx.x;
    if (e >= E + N) return;
    int s, d;
    edge_nodes(src, dst, E, e, s, d);
    for (int h = 0; h < H; ++h) {
        const float v  = leaky(a_s[(size_t)s * H + h] + a_d[(size_t)d * H + h]);
        const float ex = __expf(v - m[(size_t)d * H + h]);
        atomicAdd(&denom[(size_t)d * H + h], ex);
    }
}

// --------------------------- weighted scatter-add --------------------------
// 16 threads per edge, each owning 4 consecutive channels (float4 gather,
// 4 global_atomic_add_f32). F = H*C, F % 4 == 0, C % 4 == 0.
__global__ void edge_scatter(const int* __restrict__ src, const int* __restrict__ dst,
                             int E, int N, int H, int C,
                             const float* __restrict__ a_s, const float* __restrict__ a_d,
                             const float* __restrict__ m, const float* __restrict__ denom,
                             const float* __restrict__ xW, float* __restrict__ accum)
{
    const int F = H * C;
    const int parts = F >> 2;
    const long long t = (long long)blockIdx.x * blockDim.x + threadIdx.x;
    const long long e = t / parts;
    if (e >= (long long)E + N) return;
    const int part = (int)(t - e * parts);
    int s, d;
    edge_nodes(src, dst, E, (int)e, s, d);

    const int c0 = part * 4;
    const int h  = c0 / C;
    const float v     = leaky(a_s[(size_t)s * H + h] + a_d[(size_t)d * H + h]);
    const float alpha = __expf(v - m[(size_t)d * H + h]) / denom[(size_t)d * H + h];

    const float4 xv = *(const float4*)(xW + (size_t)s * F + c0);
    float* ap = accum + (size_t)d * F + c0;
    atomicAdd(ap + 0, alpha * xv.x);
    atomicAdd(ap + 1, alpha * xv.y);
    atomicAdd(ap + 2, alpha * xv.z);
    atomicAdd(ap + 3, alpha * xv.w);
}

// ------------------------------ bias + ELU ---------------------------------
__global__ void finalize_elu(float* __restrict__ buf, const float* __restrict__ bias,
                             int N, int F)
{
    const int t = blockIdx.x * blockDim.x + threadIdx.x;
    if (t >= N * F) return;
    const float v = buf[t] + bias[t % F];
    buf[t] = v > 0.f ? v : (__expf(v) - 1.0f);
}

// ---------------------------------------------------------------------------
extern "C" void kernel_launch(void* const* d_in, const int* in_sizes, int n_in,
                              void* d_out, int out_size, void* d_ws, size_t ws_size,
                              hipStream_t stream)
{
    const float* x        = (const float*)d_in[0];
    const int*   eidx     = (const int*)  d_in[1];
    const float* W0       = (const float*)d_in[2];
    const float* att_src0 = (const float*)d_in[3];
    const float* att_dst0 = (const float*)d_in[4];
    const float* b0       = (const float*)d_in[5];
    const float* W1       = (const float*)d_in[6];
    const float* att_src1 = (const float*)d_in[7];
    const float* att_dst1 = (const float*)d_in[8];
    const float* b1       = (const float*)d_in[9];

    const int IN = 128, F = 64, H0 = 8, C0 = 8;
    const int Nn = in_sizes[0] / IN;          // 100000 (multiple of 16)
    const int E  = in_sizes[1] / 2;           // 1.6M
    const int Et = E + Nn;                    // + self loops
    const int* src = eidx;
    const int* dst = eidx + E;

    // workspace layout (floats)
    float* ws   = (float*)d_ws;
    float* xW   = ws;                         // Nn*F   (xW0, then xW1)
    float* hbuf = xW   + (size_t)Nn * F;      // Nn*F   (accum0 -> h)
    float* a_s  = hbuf + (size_t)Nn * F;      // Nn*H0
    float* a_d  = a_s  + (size_t)Nn * H0;     // Nn*H0
    float* mbuf = a_d  + (size_t)Nn * H0;     // Nn*H0
    float* dbuf = mbuf + (size_t)Nn * H0;     // Nn*H0
    float* out  = (float*)d_out;              // Nn*F (accum1 in place)

    const dim3 blk(256);
    auto nb = [](long long t) { return dim3((unsigned)((t + 255) / 256)); };
    const int strips = Nn / 16;               // 6250 waves
    const dim3 gemmg((strips + 7) / 8);       // 8 waves / block

    // ----------------------------- layer 0 --------------------------------
    gemm_wmma_f32_strip<<<gemmg, blk, 0, stream>>>(x, W0, xW, Nn, IN);
    node_attn<<<nb((long long)Nn * H0), blk, 0, stream>>>(xW, att_src0, att_dst0,
                                                          Nn, H0, C0, a_s, a_d);
    fill_f32<<<nb((long long)Nn * H0), blk, 0, stream>>>(mbuf, -INFINITY, Nn * H0);
    fill_f32<<<nb((long long)Nn * H0), blk, 0, stream>>>(dbuf, 0.0f, Nn * H0);
    fill_f32<<<nb((long long)Nn * F),  blk, 0, stream>>>(hbuf, 0.0f, Nn * F);
    edge_max<<<nb(Et), blk, 0, stream>>>(src, dst, E, Nn, H0, a_s, a_d, mbuf);
    edge_sum<<<nb(Et), blk, 0, stream>>>(src, dst, E, Nn, H0, a_s, a_d, mbuf, dbuf);
    edge_scatter<<<nb((long long)Et * (F / 4)), blk, 0, stream>>>(
        src, dst, E, Nn, H0, C0, a_s, a_d, mbuf, dbuf, xW, hbuf);
    finalize_elu<<<nb((long long)Nn * F), blk, 0, stream>>>(hbuf, b0, Nn, F);

    // ----------------------------- layer 1 --------------------------------
    gemm_wmma_f32_strip<<<gemmg, blk, 0, stream>>>(hbuf, W1, xW, Nn, F);
    node_attn<<<nb(Nn), blk, 0, stream>>>(xW, att_src1, att_dst1, Nn, 1, F, a_s, a_d);
    fill_f32<<<nb(Nn), blk, 0, stream>>>(mbuf, -INFINITY, Nn);
    fill_f32<<<nb(Nn), blk, 0, stream>>>(dbuf, 0.0f, Nn);
    fill_f32<<<nb((long long)Nn * F), blk, 0, stream>>>(out, 0.0f, Nn * F);
    edge_max<<<nb(Et), blk, 0, stream>>>(src, dst, E, Nn, 1, a_s, a_d, mbuf);
    edge_sum<<<nb(Et), blk, 0, stream>>>(src, dst, E, Nn, 1, a_s, a_d, mbuf, dbuf);
    edge_scatter<<<nb((long long)Et * (F / 4)), blk, 0, stream>>>(
        src, dst, E, Nn, 1, F, a_s, a_d, mbuf, dbuf, xW, out);
    finalize_elu<<<nb((long long)Nn * F), blk, 0, stream>>>(out, b1, Nn, F);

    (void)n_in; (void)out_size; (void)ws_size;
}